// LinearAttention_29257317220999
// MI455X (gfx1250) — compile-verified
//
#include <hip/hip_runtime.h>

#define L_SEQ  4096
#define DK     128
#define DV     128
#define CHUNK  16
#define NSEG   16
#define SEG_LEN    (L_SEQ / NSEG)     // 256
#define SEG_CHUNKS (SEG_LEN / CHUNK)  // 16

typedef __attribute__((ext_vector_type(2))) float v2f;
typedef __attribute__((ext_vector_type(8))) float v8f;

// D(16x16,f32) = A(16x4,f32) x B(4x16,f32) + C   -- full-precision WMMA
__device__ __forceinline__ v8f wmma_f32(v2f a, v2f b, v8f c) {
  return __builtin_amdgcn_wmma_f32_16x16x4_f32(
      /*neg_a=*/false, a, /*neg_b=*/false, b,
      /*c_mod=*/(short)0, c, /*reuse_a=*/false, /*reuse_b=*/false);
}

// elu(x)+1 : positive feature map
__device__ __forceinline__ float phi_map(float x) {
  return x > 0.0f ? x + 1.0f : __expf(x);
}

// Shared-memory tile set (one wave per workgroup).
struct SmemT {
  float sQ[16][132];   // phi(Q) chunk (+4 pad for bank spread)
  float sK[16][132];   // phi(K) chunk
  float sV[16][20];    // V tile (seq x dv16)
  float sSc[16][20];   // masked scores
  float sSv[128][20];  // state: sum phiK^T V  (dk x dv16)
  float sSz[128][20];  // state: sum phiK^T 1  (dk x 16, cols equal)
};

__device__ __forceinline__ void zero_state(SmemT& sm, int lane) {
  for (int i = lane; i < 128 * 20; i += 32) {
    (&sm.sSv[0][0])[i] = 0.0f;
    (&sm.sSz[0][0])[i] = 0.0f;
  }
}

__device__ __forceinline__ void stage_chunk_Q(SmemT& sm, const float4* Q4,
                                              int rowbase, int lane) {
  #pragma unroll
  for (int i = 0; i < 16; ++i) {
    int f = lane + 32 * i;   // 0..511 float4 slots (16 rows x 32 f4)
    int r = f >> 5, c4 = f & 31;
    float4 q = Q4[(size_t)(rowbase + r) * 32 + c4];
    sm.sQ[r][4 * c4 + 0] = phi_map(q.x);
    sm.sQ[r][4 * c4 + 1] = phi_map(q.y);
    sm.sQ[r][4 * c4 + 2] = phi_map(q.z);
    sm.sQ[r][4 * c4 + 3] = phi_map(q.w);
  }
}

__device__ __forceinline__ void stage_chunk_KV(SmemT& sm, const float4* K4,
                                               const float4* V4, int rowbase,
                                               int dv0, int lane) {
  #pragma unroll
  for (int i = 0; i < 16; ++i) {
    int f = lane + 32 * i;
    int r = f >> 5, c4 = f & 31;
    float4 k = K4[(size_t)(rowbase + r) * 32 + c4];
    sm.sK[r][4 * c4 + 0] = phi_map(k.x);
    sm.sK[r][4 * c4 + 1] = phi_map(k.y);
    sm.sK[r][4 * c4 + 2] = phi_map(k.z);
    sm.sK[r][4 * c4 + 3] = phi_map(k.w);
  }
  #pragma unroll
  for (int i = 0; i < 2; ++i) {
    int f = lane + 32 * i;   // 0..63 float4 slots (16 rows x 4 f4)
    int r = f >> 2, c4 = f & 3;
    float4 v = V4[(size_t)(rowbase + r) * 32 + (dv0 >> 2) + c4];
    sm.sV[r][4 * c4 + 0] = v.x;
    sm.sV[r][4 * c4 + 1] = v.y;
    sm.sV[r][4 * c4 + 2] = v.z;
    sm.sV[r][4 * c4 + 3] = v.w;
  }
}

// State update: S_t += phiK_t^T(16x16) @ [V | 1]  for all 8 dk tiles.
__device__ __forceinline__ void state_update(SmemT& sm, int m16, int hi) {
  const v2f bones = {1.0f, 1.0f};
  for (int t = 0; t < 8; ++t) {
    int dk0 = 16 * t;
    v8f cv, cz;
    #pragma unroll
    for (int p = 0; p < 8; ++p) {
      int mm = dk0 + p + 8 * hi;
      cv[p] = sm.sSv[mm][m16];
      cz[p] = sm.sSz[mm][m16];
    }
    #pragma unroll
    for (int kk = 0; kk < 4; ++kk) {
      int kb = 4 * kk + 2 * hi;                                  // seq index j
      v2f a  = {sm.sK[kb][dk0 + m16], sm.sK[kb + 1][dk0 + m16]}; // phiK^T
      v2f bv = {sm.sV[kb][m16], sm.sV[kb + 1][m16]};
      cv = wmma_f32(a, bv, cv);
      cz = wmma_f32(a, bones, cz);
    }
    #pragma unroll
    for (int p = 0; p < 8; ++p) {
      int mm = dk0 + p + 8 * hi;
      sm.sSv[mm][m16] = cv[p];
      sm.sSz[mm][m16] = cz[p];
    }
  }
}

// Forward for one chunk: scores, causal mask, O = Sc@[V|1] + phiQ@[Sv|Sz],
// normalize and store. Uses OLD state (call before state_update).
__device__ __forceinline__ void forward_chunk(SmemT& sm, float* __restrict__ Out,
                                              int rowbase, int dv0, int m16,
                                              int hi) {
  const v2f bones = {1.0f, 1.0f};
  v8f sc = {0.f, 0.f, 0.f, 0.f, 0.f, 0.f, 0.f, 0.f};
  for (int kk = 0; kk < 32; ++kk) {
    int kb = 4 * kk + 2 * hi;
    v2f a    = {sm.sQ[m16][kb], sm.sQ[m16][kb + 1]};   // A[m][k] = phiQ[m][k]
    v2f bmat = {sm.sK[m16][kb], sm.sK[m16][kb + 1]};   // B[k][n] = phiK[n][k]
    sc = wmma_f32(a, bmat, sc);
  }
  #pragma unroll
  for (int p = 0; p < 8; ++p) {
    int m = p + 8 * hi;
    sm.sSc[m][m16] = (m16 <= m) ? sc[p] : 0.0f;        // causal mask
  }
  __syncthreads();

  v8f ov = {0.f, 0.f, 0.f, 0.f, 0.f, 0.f, 0.f, 0.f};
  v8f oz = {0.f, 0.f, 0.f, 0.f, 0.f, 0.f, 0.f, 0.f};
  #pragma unroll
  for (int kk = 0; kk < 4; ++kk) {
    int kb = 4 * kk + 2 * hi;
    v2f a  = {sm.sSc[m16][kb], sm.sSc[m16][kb + 1]};
    v2f bv = {sm.sV[kb][m16], sm.sV[kb + 1][m16]};
    ov = wmma_f32(a, bv, ov);
    oz = wmma_f32(a, bones, oz);
  }
  for (int kk = 0; kk < 32; ++kk) {
    int kb = 4 * kk + 2 * hi;
    v2f a  = {sm.sQ[m16][kb], sm.sQ[m16][kb + 1]};
    v2f bv = {sm.sSv[kb][m16], sm.sSv[kb + 1][m16]};
    v2f bz = {sm.sSz[kb][m16], sm.sSz[kb + 1][m16]};
    ov = wmma_f32(a, bv, ov);
    oz = wmma_f32(a, bz, oz);
  }
  #pragma unroll
  for (int p = 0; p < 8; ++p) {
    int m = p + 8 * hi;
    Out[(size_t)(rowbase + m) * DV + dv0 + m16] = ov[p] / (oz[p] + 1e-6f);
  }
}

// ---------------- Phase 1: per-segment partial states ----------------
__global__ __launch_bounds__(32)
void linatt_phase1_partial(const float* __restrict__ K, const float* __restrict__ V,
                           float* __restrict__ wsSv, float* __restrict__ wsZ) {
  __shared__ SmemT sm;
  const int lane = threadIdx.x, m16 = lane & 15, hi = lane >> 4;
  const int idx = blockIdx.x;               // (b*8+vtile)*NSEG + seg
  const int seg = idx % NSEG;
  const int bv = idx / NSEG;
  const int vtile = bv & 7, batch = bv >> 3;
  const int dv0 = vtile * 16;

  zero_state(sm, lane);
  __syncthreads();

  const float4* K4 = reinterpret_cast<const float4*>(K);
  const float4* V4 = reinterpret_cast<const float4*>(V);

  for (int c = 0; c < SEG_CHUNKS; ++c) {
    const int rowbase = batch * L_SEQ + seg * SEG_LEN + c * CHUNK;
    stage_chunk_KV(sm, K4, V4, rowbase, dv0, lane);
    __syncthreads();
    state_update(sm, m16, hi);
    __syncthreads();
  }
  // write partial state (128x16) and, for vtile 0, the Z partial (128)
  for (int i = lane; i < 2048; i += 32)
    wsSv[(size_t)idx * 2048 + i] = sm.sSv[i >> 4][i & 15];
  if (vtile == 0)
    for (int i = lane; i < 128; i += 32)
      wsZ[(size_t)(batch * NSEG + seg) * 128 + i] = sm.sSz[i][0];
}

// ---------------- Phase 2: exclusive prefix over segments ----------------
__global__ __launch_bounds__(256)
void linatt_phase2_prefix(float* __restrict__ wsSv, float* __restrict__ wsZ) {
  const int bv = blockIdx.x;                // b*8+vtile
  const int t = threadIdx.x;
  for (int e = t; e < 2048; e += 256) {
    float run = 0.0f;
    for (int s = 0; s < NSEG; ++s) {
      size_t off = ((size_t)bv * NSEG + s) * 2048 + e;
      float v = wsSv[off];
      wsSv[off] = run;
      run += v;
    }
  }
  if ((bv & 7) == 0) {
    const int batch = bv >> 3;
    for (int e = t; e < 128; e += 256) {
      float run = 0.0f;
      for (int s = 0; s < NSEG; ++s) {
        size_t off = ((size_t)batch * NSEG + s) * 128 + e;
        float v = wsZ[off];
        wsZ[off] = run;
        run += v;
      }
    }
  }
}

// ---------------- Phase 3: seeded local scan + outputs ----------------
__global__ __launch_bounds__(32)
void linatt_phase3_scan(const float* __restrict__ Q, const float* __restrict__ K,
                        const float* __restrict__ V, float* __restrict__ Out,
                        const float* __restrict__ wsSv,
                        const float* __restrict__ wsZ) {
  __shared__ SmemT sm;
  const int lane = threadIdx.x, m16 = lane & 15, hi = lane >> 4;
  const int idx = blockIdx.x;
  const int seg = idx % NSEG;
  const int bv = idx / NSEG;
  const int vtile = bv & 7, batch = bv >> 3;
  const int dv0 = vtile * 16;

  // seed state from exclusive prefix
  for (int i = lane; i < 2048; i += 32) {
    sm.sSv[i >> 4][i & 15] = wsSv[(size_t)idx * 2048 + i];
    sm.sSz[i >> 4][i & 15] = wsZ[(size_t)(batch * NSEG + seg) * 128 + (i >> 4)];
  }
  __syncthreads();

  const float4* Q4 = reinterpret_cast<const float4*>(Q);
  const float4* K4 = reinterpret_cast<const float4*>(K);
  const float4* V4 = reinterpret_cast<const float4*>(V);

  for (int c = 0; c < SEG_CHUNKS; ++c) {
    const int rowbase = batch * L_SEQ + seg * SEG_LEN + c * CHUNK;
    stage_chunk_Q(sm, Q4, rowbase, lane);
    stage_chunk_KV(sm, K4, V4, rowbase, dv0, lane);
    __syncthreads();
    forward_chunk(sm, Out, rowbase, dv0, m16, hi);
    __syncthreads();
    state_update(sm, m16, hi);
    __syncthreads();
  }
}

// ---------------- Fallback: monolithic scan (if ws too small) ----------------
__global__ __launch_bounds__(32)
void linatt_monolithic(const float* __restrict__ Q, const float* __restrict__ K,
                       const float* __restrict__ V, float* __restrict__ Out) {
  __shared__ SmemT sm;
  const int lane = threadIdx.x, m16 = lane & 15, hi = lane >> 4;
  const int vtile = blockIdx.x & 7, batch = blockIdx.x >> 3;
  const int dv0 = vtile * 16;

  zero_state(sm, lane);
  __syncthreads();

  const float4* Q4 = reinterpret_cast<const float4*>(Q);
  const float4* K4 = reinterpret_cast<const float4*>(K);
  const float4* V4 = reinterpret_cast<const float4*>(V);

  for (int c = 0; c < L_SEQ / CHUNK; ++c) {
    const int rowbase = batch * L_SEQ + c * CHUNK;
    stage_chunk_Q(sm, Q4, rowbase, lane);
    stage_chunk_KV(sm, K4, V4, rowbase, dv0, lane);
    __syncthreads();
    forward_chunk(sm, Out, rowbase, dv0, m16, hi);
    __syncthreads();
    state_update(sm, m16, hi);
    __syncthreads();
  }
}

extern "C" void kernel_launch(void* const* d_in, const int* in_sizes, int n_in,
                              void* d_out, int out_size, void* d_ws, size_t ws_size,
                              hipStream_t stream) {
  const float* Q = (const float*)d_in[0];
  const float* K = (const float*)d_in[1];
  const float* V = (const float*)d_in[2];
  float* Out     = (float*)d_out;

  const int B = in_sizes[0] / (L_SEQ * DK);   // = 2
  const size_t nSv = (size_t)B * 8 * NSEG * 2048;   // partial KV states
  const size_t nZ  = (size_t)B * NSEG * 128;        // partial K sums
  const size_t need = (nSv + nZ) * sizeof(float);

  if (ws_size >= need) {
    float* wsSv = (float*)d_ws;
    float* wsZ  = wsSv + nSv;
    linatt_phase1_partial<<<dim3(B * 8 * NSEG), dim3(32), 0, stream>>>(K, V, wsSv, wsZ);
    linatt_phase2_prefix<<<dim3(B * 8), dim3(256), 0, stream>>>(wsSv, wsZ);
    linatt_phase3_scan<<<dim3(B * 8 * NSEG), dim3(32), 0, stream>>>(Q, K, V, Out, wsSv, wsZ);
  } else {
    linatt_monolithic<<<dim3(B * 8), dim3(32), 0, stream>>>(Q, K, V, Out);
  }
}